// GraphAttention_27333171872092
// MI455X (gfx1250) — compile-verified
//
#include <hip/hip_runtime.h>
#include <hip/hip_bf16.h>
#include <stdint.h>

#define HIDDEN 128
#define NHEADS 8
#define HEAD_DIM 16
#define TEMP 0.08838834764831845f   // 128^-0.5 (reference scales by hidden^-0.5)
#define SLOPE 0.2f

typedef float v2f __attribute__((ext_vector_type(2)));
typedef float v4f __attribute__((ext_vector_type(4)));
typedef float v8f __attribute__((ext_vector_type(8)));

// ---------------------------------------------------------------- utilities
__global__ void k_zero_u32(uint32_t* __restrict__ p, int n) {
    int i = blockIdx.x * 256 + threadIdx.x;
    if (i < n) p[i] = 0u;
}

// ---------------------------------------------------------------- K1: scores via WMMA
// One wave = one 16-edge tile. Stage K/Q tiles (16x128 f32) into padded LDS,
// then per head: 4x V_WMMA_F32_16X16X4_F32 accumulating K_tile * Q_tile^T;
// diagonal of the 16x16 result = the 16 per-edge dot products.
struct __align__(16) WaveTile {
    float k[16][132];   // 132 = 128 + 4 pad -> conflict-free ds_load_b64 column reads
    float q[16][132];
    float sc[16][8];    // extracted diagonal scores [edge][head]
};

__global__ __launch_bounds__(64)
void k_scores(const float* __restrict__ keys, const float* __restrict__ queries,
              const int* __restrict__ dst, float* __restrict__ scores,
              uint32_t* __restrict__ deg, int E, int ntiles)
{
    __shared__ WaveTile tile[2];
    const int tid  = threadIdx.x;
    const int w    = tid >> 5;
    const int lane = tid & 31;
    const int t    = blockIdx.x * 2 + w;
    const bool active = (t < ntiles);
    const int tbase = (active ? t : (ntiles - 1)) * 16;
    WaveTile& T = tile[w];

    // ---- stage 16 rows x 128 floats of K and Q (coalesced 512B per row) ----
    for (int row = 0; row < 16; ++row) {
        long er = (long)tbase + row;
        if (er >= E) er = E - 1;                       // clamp tail (writes are guarded later)
        v4f kv = __builtin_nontemporal_load((const v4f*)(keys    + (size_t)er * HIDDEN) + lane);
        v4f qv = __builtin_nontemporal_load((const v4f*)(queries + (size_t)er * HIDDEN) + lane);
        *(v4f*)&T.k[row][lane * 4] = kv;
        *(v4f*)&T.q[row][lane * 4] = qv;
    }
    __syncthreads();

    // ---- WMMA phase: EXEC is all-ones here ----
    const int half = lane >> 4;      // 0: K-chunk {0,1}, 1: K-chunk {2,3}
    const int r16  = lane & 15;      // matrix row (A) / col (B)
    for (int h = 0; h < NHEADS; ++h) {
        v8f acc = {};
        #pragma unroll
        for (int c = 0; c < 4; ++c) {
            int col = h * HEAD_DIM + c * 4 + half * 2;
            // A[m][k] = K[m][col..col+1]  (lanes 0-15: K0,K1 ; lanes 16-31: K2,K3)
            // B[k][n] = Q[n][col..col+1]  (same per-lane indexing -> D = K * Q^T)
            v2f a = *(const v2f*)&T.k[r16][col];
            v2f b = *(const v2f*)&T.q[r16][col];
            acc = __builtin_amdgcn_wmma_f32_16x16x4_f32(
                      false, a, false, b, (short)0, acc, false, false);
        }
        // diagonal extraction: m<8 -> lane m, VGPR m ; m>=8 -> lane m+16, VGPR m-8
        float dg = 0.f;
        #pragma unroll
        for (int r = 0; r < 8; ++r) {
            float vr = acc[r];
            dg = (lane == r || lane == r + 24) ? vr : dg;
        }
        if (lane < 8)         T.sc[lane][h]      = dg;
        else if (lane >= 24)  T.sc[lane - 16][h] = dg;
    }
    __syncthreads();

    // ---- epilogue: scale + LeakyReLU, store scores, bump degree ----
    if (lane < 16) {
        long e = (long)tbase + lane;
        if (active && e < E) {
            int n = dst[e];
            float o[8];
            #pragma unroll
            for (int h = 0; h < 8; ++h) {
                float s = T.sc[lane][h] * TEMP;
                o[h] = (s >= 0.f) ? s : SLOPE * s;
            }
            v4f s0 = {o[0], o[1], o[2], o[3]};
            v4f s1 = {o[4], o[5], o[6], o[7]};
            *(v4f*)(scores + (size_t)e * 8)     = s0;
            *(v4f*)(scores + (size_t)e * 8 + 4) = s1;
            atomicAdd(&deg[n], 1u);
        }
    }
}

// ---------------------------------------------------------------- K2: exclusive scan (single block)
__global__ __launch_bounds__(1024)
void k_scan(uint32_t* __restrict__ deg, uint32_t* __restrict__ rowptr, int N)
{
    __shared__ uint32_t buf[1024];
    const int t = threadIdx.x;
    const int C = (N + 1023) >> 10;           // elements per thread

    uint32_t s = 0;
    for (int j = 0; j < C; ++j) { int i = t * C + j; if (i < N) s += deg[i]; }
    buf[t] = s;
    __syncthreads();
    for (int off = 1; off < 1024; off <<= 1) {
        uint32_t y = (t >= off) ? buf[t - off] : 0u;
        __syncthreads();
        buf[t] += y;
        __syncthreads();
    }
    uint32_t run = buf[t] - s;                // exclusive prefix of this thread's span
    for (int j = 0; j < C; ++j) {
        int i = t * C + j;
        if (i < N) {
            uint32_t d = deg[i];
            rowptr[i] = run;
            run += d;
            deg[i] = 0u;                      // re-zero: deg reused as fill cursors in K3
        }
    }
    if (t == 0) rowptr[N] = buf[1023];
}

// ---------------------------------------------------------------- K3: CSR fill
__global__ __launch_bounds__(256)
void k_fill(const int* __restrict__ dst, const uint32_t* __restrict__ rowptr,
            uint32_t* __restrict__ fill, uint32_t* __restrict__ eids, int E)
{
    int e = blockIdx.x * 256 + threadIdx.x;
    if (e < E) {
        int n = dst[e];
        uint32_t pos = atomicAdd(&fill[n], 1u);
        eids[rowptr[n] + pos] = (uint32_t)e;
    }
}

// ---------------------------------------------------------------- K4: node-major gather softmax+aggregate
// One wave per node. Lane owns 4 output dims; its head = lane>>2.
__global__ __launch_bounds__(256)
void k_gather(const float* __restrict__ values, const float* __restrict__ scores,
              const uint32_t* __restrict__ rowptr, const uint32_t* __restrict__ eids,
              float* __restrict__ out, int N)
{
    const int node = blockIdx.x * 8 + (threadIdx.x >> 5);
    const int lane = threadIdx.x & 31;
    if (node >= N) return;

    const uint32_t b  = rowptr[node];
    const uint32_t en = rowptr[node + 1];
    const int h = lane >> 2;                  // head of this lane's 4 dims

    // pass 1: per-head max (softmax stability, matches reference segment_max)
    float m = -3.4e38f;
    for (uint32_t j = b; j < en; ++j) {
        uint32_t e = eids[j];
        m = fmaxf(m, scores[(size_t)e * 8 + h]);
    }

    // pass 2: num = sum(ex * v), den = sum(ex)
    float n0 = 0.f, n1 = 0.f, n2 = 0.f, n3 = 0.f, den = 0.f;
    for (uint32_t j = b; j < en; ++j) {
        uint32_t e = eids[j];
        float s  = scores[(size_t)e * 8 + h];
        float ex = __expf(s - m);
        den += ex;
        v4f vv = __builtin_nontemporal_load((const v4f*)(values + (size_t)e * HIDDEN) + lane);
        n0 += ex * vv[0]; n1 += ex * vv[1]; n2 += ex * vv[2]; n3 += ex * vv[3];
    }

    float inv = (den > 0.f) ? (1.f / den) : 0.f;   // empty node -> zeros (matches segment_sum)
    v4f o = {n0 * inv, n1 * inv, n2 * inv, n3 * inv};
    *(v4f*)(out + (size_t)node * HIDDEN + lane * 4) = o;
}

// ---------------------------------------------------------------- launch
extern "C" void kernel_launch(void* const* d_in, const int* in_sizes, int n_in,
                              void* d_out, int out_size, void* d_ws, size_t ws_size,
                              hipStream_t stream)
{
    const float* keys    = (const float*)d_in[0];
    const float* queries = (const float*)d_in[1];
    const float* values  = (const float*)d_in[2];
    const int*   dst     = (const int*)d_in[3];
    const int E = in_sizes[3];                // n_edges
    const int N = out_size / HIDDEN;          // n_nodes
    float* out = (float*)d_out;

    // workspace carve-out (256B aligned): scores | deg | rowptr | eids  (~30 MB)
    char* ws = (char*)d_ws;
    auto carve = [&](size_t bytes) -> char* {
        char* p = ws;
        ws += (bytes + 255) & ~(size_t)255;
        return p;
    };
    float*    scores = (float*)   carve((size_t)E * 8 * sizeof(float));
    uint32_t* deg    = (uint32_t*)carve((size_t)N * sizeof(uint32_t));
    uint32_t* rowptr = (uint32_t*)carve((size_t)(N + 1) * sizeof(uint32_t));
    uint32_t* eids   = (uint32_t*)carve((size_t)E * sizeof(uint32_t));
    (void)ws_size;

    const int ntiles = (E + 15) / 16;

    k_zero_u32<<<(N + 255) / 256, 256, 0, stream>>>(deg, N);
    k_scores  <<<(ntiles + 1) / 2, 64, 0, stream>>>(keys, queries, dst, scores, deg, E, ntiles);
    k_scan    <<<1, 1024, 0, stream>>>(deg, rowptr, N);
    k_fill    <<<(E + 255) / 256, 256, 0, stream>>>(dst, rowptr, deg, eids, E);
    k_gather  <<<(N + 7) / 8, 256, 0, stream>>>(values, scores, rowptr, eids, out, N);
}